// CenterLoss_43258910605421
// MI455X (gfx1250) — compile-verified
//
#include <hip/hip_runtime.h>

typedef __attribute__((ext_vector_type(2))) float v2f;
typedef __attribute__((ext_vector_type(8))) float v8f;

#define FEAT 256
#define ALPHA 0.5f

// ---------------------------------------------------------------- zero scratch
__global__ void centerloss_zero_ws(float* __restrict__ ws, int n) {
  int i = blockIdx.x * blockDim.x + threadIdx.x;
  int stride = gridDim.x * blockDim.x;
  for (; i < n; i += stride) ws[i] = 0.0f;
}

// ------------------------------------------------- fused loss + scatter pass
// One wave per feature row (grid-strided). Each lane owns 8 columns as two
// float4 (global_load_b128). Gathered center rows are L2-resident (10 MB in
// 192 MB L2); the next row's center line is prefetched (global_prefetch_b8).
// Squared-diff partials are reduced across the wave with a single
// v_wmma_f32_16x16x4_f32 (A = per-lane float2 partials, B = ones).
__global__ __launch_bounds__(256) void centerloss_fused(
    const float* __restrict__ features,   // [B, 256]
    const int*   __restrict__ labels,     // [B]
    const float* __restrict__ centers,    // [C, 256]
    float*       __restrict__ seg_sum,    // [C, 256] scratch (zeroed)
    float*       __restrict__ counts,     // [C]      scratch (zeroed)
    double*      __restrict__ loss_acc,   // scratch (zeroed)
    int B) {
  const int lane       = threadIdx.x & 31;
  const int waveInBlk  = threadIdx.x >> 5;
  const int wavesPerBlk = blockDim.x >> 5;
  const int wave   = blockIdx.x * wavesPerBlk + waveInBlk;
  const int nWaves = gridDim.x * wavesPerBlk;

  float accx = 0.0f, accy = 0.0f;

  for (int r = wave; r < B; r += nWaves) {           // wave-uniform loop
    const int lbl = labels[r];

    // Prefetch next iteration's gathered center row (hides L2 gather latency).
    const int rn = r + nWaves;
    if (rn < B) {                                    // wave-uniform branch
      const int lbln = labels[rn];
      const char* p = (const char*)(centers + (size_t)lbln * FEAT) + lane * 32;
      __builtin_prefetch(p, 0, 0);                   // -> global_prefetch_b8
    }

    const float4* frow = (const float4*)(features + (size_t)r   * FEAT);
    const float4* crow = (const float4*)(centers  + (size_t)lbl * FEAT);
    float4 f0 = frow[lane];
    float4 f1 = frow[lane + 32];
    float4 c0 = crow[lane];
    float4 c1 = crow[lane + 32];

    float dx0 = f0.x - c0.x, dy0 = f0.y - c0.y, dz0 = f0.z - c0.z, dw0 = f0.w - c0.w;
    float dx1 = f1.x - c1.x, dy1 = f1.y - c1.y, dz1 = f1.z - c1.z, dw1 = f1.w - c1.w;
    accx += dx0*dx0 + dy0*dy0 + dz0*dz0 + dw0*dw0;
    accy += dx1*dx1 + dy1*dy1 + dz1*dz1 + dw1*dw1;

    // Scatter-add features into per-class segment sums (L2 f32 atomics).
    float* s = seg_sum + (size_t)lbl * FEAT;
    const int col = lane * 4;
    atomicAdd(&s[col + 0],       f0.x);
    atomicAdd(&s[col + 1],       f0.y);
    atomicAdd(&s[col + 2],       f0.z);
    atomicAdd(&s[col + 3],       f0.w);
    atomicAdd(&s[128 + col + 0], f1.x);
    atomicAdd(&s[128 + col + 1], f1.y);
    atomicAdd(&s[128 + col + 2], f1.z);
    atomicAdd(&s[128 + col + 3], f1.w);
    if (lane == 0) atomicAdd(&counts[lbl], 1.0f);
  }

  // ---- cross-lane reduction via matrix core: C[m,n] = sum_k A[m,k] (B = 1s)
  // A (16x4 f32) = exactly one float2 per lane; EXEC is all-1s here.
  v2f a; a.x = accx; a.y = accy;
  v2f b; b.x = 1.0f; b.y = 1.0f;
  v8f c = {};
  c = __builtin_amdgcn_wmma_f32_16x16x4_f32(
      /*neg_a=*/false, a, /*neg_b=*/false, b,
      /*c_mod=*/(short)0, c, /*reuse_a=*/false, /*reuse_b=*/false);
  // Lane's 8 C VGPRs hold one column over half the M rows; combine halves.
  float s8 = c[0] + c[1] + c[2] + c[3] + c[4] + c[5] + c[6] + c[7];
  s8 += __shfl_xor(s8, 16, 32);     // lanes 0-15 <-> 16-31: full wave total

  __shared__ float red[8];
  if (lane == 0) red[waveInBlk] = s8;
  __syncthreads();
  if (threadIdx.x == 0) {
    float t = 0.0f;
    for (int i = 0; i < wavesPerBlk; ++i) t += red[i];
    atomicAdd(loss_acc, (double)t);   // global_atomic_add_f64, once per block
  }
}

// ------------------------------------------------------------------ finalize
// out[0] = loss; out[1 + c*256 + col] = updated centers.
__global__ __launch_bounds__(256) void centerloss_finalize(
    const float*  __restrict__ seg_sum,
    const float*  __restrict__ counts,
    const float*  __restrict__ centers,
    const double* __restrict__ loss_acc,
    float*        __restrict__ out,
    long long BD) {
  const int cls = blockIdx.x;
  const int col = threadIdx.x;
  if (cls == 0 && col == 0)
    out[0] = (float)(*loss_acc / (double)BD);

  const float cnt = counts[cls];                 // broadcast within block
  const size_t idx = (size_t)cls * FEAT + col;
  const float c0 = centers[idx];
  float o = c0;
  if (cnt > 0.0f) {
    const float mean = seg_sum[idx] / cnt;
    o = c0 + ALPHA * (mean - c0);
  }
  out[1 + idx] = o;
}

// ---------------------------------------------------------------------- host
extern "C" void kernel_launch(void* const* d_in, const int* in_sizes, int n_in,
                              void* d_out, int out_size, void* d_ws, size_t ws_size,
                              hipStream_t stream) {
  const float* features = (const float*)d_in[0];
  const int*   labels   = (const int*)d_in[1];
  const float* centers  = (const float*)d_in[2];

  const int B = in_sizes[1];              // 131072
  const int D = FEAT;                     // 256
  const int C = in_sizes[2] / D;          // 10000

  // Workspace layout (aligned): [double lossAcc][counts C][seg_sum C*D]
  char*   ws       = (char*)d_ws;
  double* loss_acc = (double*)ws;
  size_t  countsOff = 256;
  size_t  segOff    = countsOff + (((size_t)C * 4 + 255) / 256) * 256;
  float*  counts   = (float*)(ws + countsOff);
  float*  seg_sum  = (float*)(ws + segOff);
  int zeroN = (int)((segOff + (size_t)C * D * 4) / 4);

  centerloss_zero_ws<<<1024, 256, 0, stream>>>((float*)ws, zeroN);
  centerloss_fused<<<2048, 256, 0, stream>>>(features, labels, centers,
                                             seg_sum, counts, loss_acc, B);
  centerloss_finalize<<<C, 256, 0, stream>>>(seg_sum, counts, centers, loss_acc,
                                             (float*)d_out, (long long)B * D);
}